// QuantumTextClassifier_70377334112283
// MI455X (gfx1250) — compile-verified
//
#include <hip/hip_runtime.h>
#include <math.h>

typedef __attribute__((ext_vector_type(2))) float v2f;
typedef __attribute__((ext_vector_type(4))) float f4v;
typedef __attribute__((ext_vector_type(8))) float v8f;

#define WAVES 4
#define TILE  (WAVES * 16)
#define ISTR  68   // padded LDS row stride (floats) -> bank-conflict-free

// ---------------------------------------------------------------------------
// Kernel 0: build the (transposed) fixed 64x64 circuit unitary from qparams.
// Thread c evolves basis state |c> through all gates; stores row c of U^T
// (i.e. Ut[k=c][s] = U[s][c]) split into real/imag planes in the workspace.
// Qubit q <-> bit (5-q) of the state index (qubit 0 is the outermost axis).
// ---------------------------------------------------------------------------
__global__ __launch_bounds__(64) void build_unitary_kernel(
    const float* __restrict__ qp, float* __restrict__ Ut_r, float* __restrict__ Ut_i) {
  __shared__ float vr[64][64];
  __shared__ float vi[64][64];
  const int c = threadIdx.x;
  float* rv = vr[c];
  float* iv = vi[c];
  for (int s = 0; s < 64; ++s) { rv[s] = (s == c) ? 1.0f : 0.0f; iv[s] = 0.0f; }

  auto cnot = [&](int cq, int tq) {
    const int mc = 1 << (5 - cq), mt = 1 << (5 - tq);
    for (int s = 0; s < 64; ++s) {
      if ((s & mc) && !(s & mt)) {
        const int s1 = s | mt;
        float t = rv[s]; rv[s] = rv[s1]; rv[s1] = t;
        t = iv[s]; iv[s] = iv[s1]; iv[s1] = t;
      }
    }
  };
  auto rx = [&](int q, float th) {  // exp(-i th/2 X)
    float sh, ch; sincosf(0.5f * th, &sh, &ch);
    const int m = 1 << (5 - q);
    for (int s0 = 0; s0 < 64; ++s0) if (!(s0 & m)) {
      const int s1 = s0 | m;
      const float r0 = rv[s0], i0 = iv[s0], r1 = rv[s1], i1 = iv[s1];
      rv[s0] = ch * r0 + sh * i1;  iv[s0] = ch * i0 - sh * r1;
      rv[s1] = ch * r1 + sh * i0;  iv[s1] = ch * i1 - sh * r0;
    }
  };
  auto ry = [&](int q, float th) {  // exp(-i th/2 Y) = [[c,-s],[s,c]]
    float sh, ch; sincosf(0.5f * th, &sh, &ch);
    const int m = 1 << (5 - q);
    for (int s0 = 0; s0 < 64; ++s0) if (!(s0 & m)) {
      const int s1 = s0 | m;
      const float r0 = rv[s0], i0 = iv[s0], r1 = rv[s1], i1 = iv[s1];
      rv[s0] = ch * r0 - sh * r1;  iv[s0] = ch * i0 - sh * i1;
      rv[s1] = sh * r0 + ch * r1;  iv[s1] = sh * i0 + ch * i1;
    }
  };
  auto rz = [&](int q, float th) {  // diag(e^{-i th/2}, e^{+i th/2})
    float sh, ch; sincosf(0.5f * th, &sh, &ch);
    const int m = 1 << (5 - q);
    for (int s0 = 0; s0 < 64; ++s0) if (!(s0 & m)) {
      const int s1 = s0 | m;
      const float r0 = rv[s0], i0 = iv[s0], r1 = rv[s1], i1 = iv[s1];
      rv[s0] = ch * r0 + sh * i0;  iv[s0] = ch * i0 - sh * r0;
      rv[s1] = ch * r1 - sh * i1;  iv[s1] = ch * i1 + sh * r1;
    }
  };
  auto crz = [&](int cq, int tq, float th) {  // phase on control==1
    float sh, ch; sincosf(0.5f * th, &sh, &ch);
    const int mc = 1 << (5 - cq), mt = 1 << (5 - tq);
    for (int s = 0; s < 64; ++s) if (s & mc) {
      const float r = rv[s], i = iv[s];
      if (s & mt) { rv[s] = ch * r - sh * i; iv[s] = ch * i + sh * r; }   // e^{+i th/2}
      else        { rv[s] = ch * r + sh * i; iv[s] = ch * i - sh * r; }  // e^{-i th/2}
    }
  };

  for (int layer = 0; layer < 3; ++layer) {
    cnot(0, 1); cnot(2, 3); cnot(4, 5);
    cnot(1, 2); cnot(3, 4);
    for (int i = 0; i < 6; ++i) {
      const float tx = qp[layer * 18 + i * 3 + 0];
      const float ty = qp[layer * 18 + i * 3 + 1];
      const float tz = qp[layer * 18 + i * 3 + 2];
      rx(i, tx); ry(i, ty); rz(i, tz);
    }
    if (layer < 2)
      for (int i = 0; i < 6; ++i)
        crz(i, (i + 1) % 6, qp[layer * 18 + i * 3 + 0]);
  }
  for (int s = 0; s < 64; ++s) { Ut_r[c * 64 + s] = rv[s]; Ut_i[c * 64 + s] = iv[s]; }
}

// ---------------------------------------------------------------------------
// Main kernel: MLP prologue -> real product state -> complex 64x64 matvec via
// f32 WMMA (psi_r = init@Ut_r, psi_i = init@Ut_i) -> probs -> <Z_q> -> MLP.
// ---------------------------------------------------------------------------
__global__ __launch_bounds__(WAVES * 32) void qnn_kernel(
    const float* __restrict__ x,
    const float* __restrict__ W1, const float* __restrict__ b1,
    const float* __restrict__ W2, const float* __restrict__ b2,
    const float* __restrict__ W3, const float* __restrict__ b3,
    const float* __restrict__ W4, const float* __restrict__ b4,
    const float* __restrict__ Ut_r, const float* __restrict__ Ut_i,
    float* __restrict__ out, int batch) {
  __shared__ float sUr[64 * 64];
  __shared__ float sUi[64 * 64];
  __shared__ float sAmp[WAVES][16 * ISTR];

  const int tid  = threadIdx.x;
  const int wave = tid >> 5;
  const int lane = tid & 31;
  const int half = lane >> 4;
  const int l    = lane & 15;
  const int ebase = blockIdx.x * TILE + wave * 16;

  // Stage U^T (real+imag) into LDS, 128-bit vectorized.
  {
    const f4v* srcR = (const f4v*)Ut_r;
    const f4v* srcI = (const f4v*)Ut_i;
    f4v* dR = (f4v*)sUr;
    f4v* dI = (f4v*)sUi;
    for (int i = tid; i < 1024; i += WAVES * 32) { dR[i] = srcR[i]; dI[i] = srcI[i]; }
  }

  // Prologue: lanes 0..15 each own one batch element of this wave's tile.
  if (lane < 16) {
    int e = ebase + lane;
    const int ec = (e < batch) ? e : (batch - 1);
    const float x0 = x[ec * 4 + 0], x1 = x[ec * 4 + 1];
    const float x2 = x[ec * 4 + 2], x3 = x[ec * 4 + 3];
    float h[16];
#pragma unroll
    for (int j = 0; j < 16; ++j) {
      float t = b1[j] + x0 * W1[j] + x1 * W1[16 + j] + x2 * W1[32 + j] + x3 * W1[48 + j];
      h[j] = t > 0.0f ? t : 0.0f;
    }
    float A0[6], A1[6];
#pragma unroll
    for (int q = 0; q < 6; ++q) {
      float t = b2[q];
#pragma unroll
      for (int j = 0; j < 16; ++j) t += h[j] * W2[j * 6 + q];
      const float ang = 0.5f * 3.14159265358979323846f * tanhf(t);  // half * (pi*pre)
      float sn, cs; sincosf(ang, &sn, &cs);
      const float is2 = 0.70710678118654752440f;
      A0[q] = (cs - sn) * is2;
      A1[q] = (cs + sn) * is2;
    }
    float m01[4], m23[4], m45[4];
#pragma unroll
    for (int i = 0; i < 4; ++i) {
      m01[i] = ((i & 2) ? A1[0] : A0[0]) * ((i & 1) ? A1[1] : A0[1]);
      m23[i] = ((i & 2) ? A1[2] : A0[2]) * ((i & 1) ? A1[3] : A0[3]);
      m45[i] = ((i & 2) ? A1[4] : A0[4]) * ((i & 1) ? A1[5] : A0[5]);
    }
    float* dst = &sAmp[wave][lane * ISTR];
#pragma unroll
    for (int s = 0; s < 64; ++s)
      dst[s] = m01[s >> 4] * m23[(s >> 2) & 3] * m45[s & 3];
  }
  __syncthreads();

  // GEMM: psi_r/psi_i (16 x 64) for this wave's 16 elements.
  // A layout (16x4 f32): vgpr0 = K(0|2), vgpr1 = K(1|3); M = lane&15.
  // B layout (4x16):     vgpr0 = K(0|2), vgpr1 = K(1|3); N = lane&15.
  v8f accr[4] = {v8f{}, v8f{}, v8f{}, v8f{}};
  v8f acci[4] = {v8f{}, v8f{}, v8f{}, v8f{}};
  const float* amp = &sAmp[wave][l * ISTR];
#pragma unroll
  for (int kt = 0; kt < 16; ++kt) {
    const int kb = 4 * kt + 2 * half;
    v2f a;
    a.x = amp[kb];
    a.y = amp[kb + 1];
#pragma unroll
    for (int nt = 0; nt < 4; ++nt) {
      const int n = nt * 16 + l;
      v2f br, bi;
      br.x = sUr[kb * 64 + n];        br.y = sUr[(kb + 1) * 64 + n];
      bi.x = sUi[kb * 64 + n];        bi.y = sUi[(kb + 1) * 64 + n];
      accr[nt] = __builtin_amdgcn_wmma_f32_16x16x4_f32(
          false, a, false, br, (short)0, accr[nt], false, false);
      acci[nt] = __builtin_amdgcn_wmma_f32_16x16x4_f32(
          false, a, false, bi, (short)0, acci[nt], false, false);
    }
  }

  // Probabilities back into this wave's LDS region per the C/D layout:
  // vgpr r: lanes 0-15 -> (M=r, N=lane), lanes 16-31 -> (M=r+8, N=lane-16).
  {
    float* wamp = sAmp[wave];
#pragma unroll
    for (int nt = 0; nt < 4; ++nt)
#pragma unroll
      for (int r = 0; r < 8; ++r) {
        const float pr = accr[nt][r], pi = acci[nt][r];
        wamp[(r + 8 * half) * ISTR + nt * 16 + l] = pr * pr + pi * pi;
      }
  }
  __syncthreads();

  // Epilogue: signed reductions -> z[6]; then relu(z@W3+b3)@W4+b4.
  if (lane < 16) {
    const int e = ebase + lane;
    if (e < batch) {
      const float* p = &sAmp[wave][lane * ISTR];
      float z0 = 0, z1 = 0, z2 = 0, z3 = 0, z4 = 0, z5 = 0;
#pragma unroll
      for (int s = 0; s < 64; ++s) {
        const float v = p[s];
        z0 += (s & 32) ? -v : v;  // qubit 0 <-> bit 5
        z1 += (s & 16) ? -v : v;
        z2 += (s & 8)  ? -v : v;
        z3 += (s & 4)  ? -v : v;
        z4 += (s & 2)  ? -v : v;
        z5 += (s & 1)  ? -v : v;
      }
      const float z[6] = {z0, z1, z2, z3, z4, z5};
      float o0 = b4[0], o1 = b4[1], o2 = b4[2], o3 = b4[3];
#pragma unroll
      for (int j = 0; j < 16; ++j) {
        float g = b3[j];
#pragma unroll
        for (int q = 0; q < 6; ++q) g += z[q] * W3[q * 16 + j];
        g = g > 0.0f ? g : 0.0f;
        o0 += g * W4[j * 4 + 0]; o1 += g * W4[j * 4 + 1];
        o2 += g * W4[j * 4 + 2]; o3 += g * W4[j * 4 + 3];
      }
      float* op = out + (size_t)e * 4;
      op[0] = o0; op[1] = o1; op[2] = o2; op[3] = o3;
    }
  }
}

// ---------------------------------------------------------------------------
extern "C" void kernel_launch(void* const* d_in, const int* in_sizes, int n_in,
                              void* d_out, int out_size, void* d_ws, size_t ws_size,
                              hipStream_t stream) {
  const float* x  = (const float*)d_in[0];
  const float* W1 = (const float*)d_in[1];
  const float* b1 = (const float*)d_in[2];
  const float* W2 = (const float*)d_in[3];
  const float* b2 = (const float*)d_in[4];
  const float* qp = (const float*)d_in[5];
  const float* W3 = (const float*)d_in[6];
  const float* b3 = (const float*)d_in[7];
  const float* W4 = (const float*)d_in[8];
  const float* b4 = (const float*)d_in[9];
  float* out  = (float*)d_out;
  float* Ut_r = (float*)d_ws;          // 64*64 floats
  float* Ut_i = Ut_r + 64 * 64;        // 64*64 floats (32 KB total)

  const int batch = in_sizes[0] / 4;
  build_unitary_kernel<<<dim3(1), dim3(64), 0, stream>>>(qp, Ut_r, Ut_i);
  const int tiles = (batch + TILE - 1) / TILE;
  qnn_kernel<<<dim3(tiles), dim3(WAVES * 32), 0, stream>>>(
      x, W1, b1, W2, b2, W3, b3, W4, b4, Ut_r, Ut_i, out, batch);
}